// RPNPostProcessor_63436666962411
// MI455X (gfx1250) — compile-verified
//
#include <hip/hip_runtime.h>
#include <math.h>

typedef unsigned int u32;
typedef unsigned long long u64;
typedef __attribute__((ext_vector_type(2))) float v2f;
typedef __attribute__((ext_vector_type(8))) float v8f;

#define NIMG 2
#define HW   65536      // 256*256 proposals per image
#define PRE  1024       // PRE_NMS_TOP_N
#define POST 256        // POST_NMS_TOP_N
#define NMS_T 0.7f

// ---------- sortable float helpers (monotone bijection f32 -> u32) ----------
__device__ __forceinline__ u32 f2sort(float f) {
  u32 u = __float_as_uint(f);
  return (u & 0x80000000u) ? ~u : (u | 0x80000000u);
}
__device__ __forceinline__ float sort2f(u32 u) {
  u = (u & 0x80000000u) ? (u & 0x7FFFFFFFu) : ~u;
  return __uint_as_float(u);
}

// ---------- stage 1: build 64-bit composite keys (score desc, idx asc) ------
__global__ void build_keys_kernel(const float* __restrict__ obj,
                                  u64* __restrict__ keys) {
  int t = blockIdx.x * blockDim.x + threadIdx.x;   // 0 .. 2*HW-1
  if (t >= NIMG * HW) return;
  u32 idx = (u32)(t & (HW - 1));
  keys[t] = ((u64)f2sort(obj[t]) << 32) | (u64)(0xFFFFFFFFu - idx);
}

// ---------- stage 2: global bitonic sort (descending), one step/launch ------
__global__ void bitonic_step_kernel(u64* __restrict__ keys, int k, int j) {
  int t = blockIdx.x * blockDim.x + threadIdx.x;   // 0 .. 2*HW-1
  int img = t >> 16;
  int i   = t & (HW - 1);
  int ixj = i ^ j;
  if (ixj <= i || img >= NIMG) return;
  u64* base = keys + ((u64)img << 16);
  u64 a = base[i], b = base[ixj];
  bool up = (i & k) == 0;                          // descending overall
  if (up ? (a < b) : (a > b)) { base[i] = b; base[ixj] = a; }
}

// ---------- stage 3: decode EAST regression for top-1024 --------------------
__global__ void decode_kernel(const u64* __restrict__ keys,
                              const float* __restrict__ reg,
                              float* __restrict__ boxes,
                              float* __restrict__ scores,
                              int* __restrict__ valid) {
  int img = blockIdx.x;
  int t   = threadIdx.x;                           // 0..1023
  u64 kk  = keys[((u64)img << 16) + t];
  u32 idx = 0xFFFFFFFFu - (u32)(kk & 0xFFFFFFFFu);
  float score = sort2f((u32)(kk >> 32));
  const float* R = reg + (size_t)img * 5 * HW;
  float rt = R[0 * HW + idx] * 1024.0f;
  float rr = R[1 * HW + idx] * 1024.0f;
  float rb = R[2 * HW + idx] * 1024.0f;
  float rl = R[3 * HW + idx] * 1024.0f;
  float ang = (R[4 * HW + idx] - 0.5f) * 90.0f;
  float w = rl + rr, h = rt + rb;
  int x = idx & 255, y = idx >> 8;
  float xc = (float)x * 4.0f + 0.5f * (rr - rl);   // /SCALE = *4
  float yc = (float)y * 4.0f + 0.5f * (rb - rt);
  xc = fminf(fmaxf(xc, 0.0f), 1023.0f);            // clip_to_image
  yc = fminf(fmaxf(yc, 0.0f), 1023.0f);
  bool v = (score > 0.1f) && (w >= 4.0f) && (h >= 4.0f);
  float* B = boxes + (size_t)(img * PRE + t) * 5;
  B[0] = xc; B[1] = yc; B[2] = w; B[3] = h; B[4] = ang;
  scores[img * PRE + t] = score;
  valid[img * PRE + t] = v ? 1 : 0;
}

// ---------- stage 4a: WMMA circle quick-reject mask -------------------------
// Per wave: 16 rows x 32 cols. Pairwise dot of centers via V_WMMA_F32_16X16X4_F32
// (K=2 used), then dist^2 = |ci|^2 + |cj|^2 - 2*dot vs (Ri+Rj)^2. Ballot per
// D-row assembles half a mask word; two 16-col tiles -> full 32-bit word.
__global__ __launch_bounds__(32) void quickreject_kernel(
    const float* __restrict__ boxes, u32* __restrict__ pmask) {
  const int lane = threadIdx.x;
  const int wcol = blockIdx.x;        // 0..31 -> cols [wcol*32, wcol*32+32)
  const int mt   = blockIdx.y;        // 0..63 -> rows [mt*16, mt*16+16)
  const int img  = blockIdx.z;
  const float* B = boxes + (size_t)img * PRE * 5;

  // A operand: 16x4, K=0/1 carry (x,y) of row box, K=2/3 zero.
  int mA = mt * 16 + (lane & 15);
  const float* pA = B + mA * 5;
  v2f a;
  a.x = (lane < 16) ? pA[0] : 0.0f;
  a.y = (lane < 16) ? pA[1] : 0.0f;

  // B operands for two 16-col tiles: 4x16, K=0/1 carry (x,y), K=2/3 zero.
  int n0 = wcol * 32 + (lane & 15);
  int n1 = n0 + 16;
  const float* pN0 = B + n0 * 5;
  const float* pN1 = B + n1 * 5;
  float nx0 = pN0[0], ny0 = pN0[1];
  float nr0 = 0.5f * sqrtf(pN0[2] * pN0[2] + pN0[3] * pN0[3]);
  float nx1 = pN1[0], ny1 = pN1[1];
  float nr1 = 0.5f * sqrtf(pN1[2] * pN1[2] + pN1[3] * pN1[3]);
  v2f b0, b1;
  b0.x = (lane < 16) ? nx0 : 0.0f;  b0.y = (lane < 16) ? ny0 : 0.0f;
  b1.x = (lane < 16) ? nx1 : 0.0f;  b1.y = (lane < 16) ? ny1 : 0.0f;

  v8f c = {};
  v8f d0 = __builtin_amdgcn_wmma_f32_16x16x4_f32(false, a, false, b0,
                                                 (short)0, c, false, false);
  v8f d1 = __builtin_amdgcn_wmma_f32_16x16x4_f32(false, a, false, b1,
                                                 (short)0, c, false, false);

  // Per-lane row data: D VGPR r holds M = mt*16 + r (+8 for upper half-wave).
  int mbase = mt * 16 + ((lane >= 16) ? 8 : 0);
  float nn0 = nx0 * nx0 + ny0 * ny0;
  float nn1 = nx1 * nx1 + ny1 * ny1;
#pragma unroll
  for (int r = 0; r < 8; ++r) {
    const float* pm = B + (mbase + r) * 5;
    float mx = pm[0], my = pm[1];
    float mr = 0.5f * sqrtf(pm[2] * pm[2] + pm[3] * pm[3]);
    float mm = mx * mx + my * my;
    float rad0 = mr + nr0, rad1 = mr + nr1;
    bool p0 = (mm + nn0 - 2.0f * d0[r]) <= rad0 * rad0 + 1.0f;  // slack
    bool p1 = (mm + nn1 - 2.0f * d1[r]) <= rad1 * rad1 + 1.0f;
    u32 w0 = __builtin_amdgcn_ballot_w32(p0);
    u32 w1 = __builtin_amdgcn_ballot_w32(p1);
    if (lane == 0) {
      int row_lo = mt * 16 + r;
      int row_hi = row_lo + 8;
      pmask[((size_t)(img * PRE + row_lo)) * 32 + wcol] =
          (w0 & 0xFFFFu) | ((w1 & 0xFFFFu) << 16);
      pmask[((size_t)(img * PRE + row_hi)) * 32 + wcol] =
          (w0 >> 16) | (w1 & 0xFFFF0000u);
    }
  }
}

// ---------- rotated-box geometry -------------------------------------------
__device__ __forceinline__ void rbox_corners_dev(const float b[5],
                                                 float cx[4], float cy[4]) {
  float th = b[4] * 0.017453292519943295f;
  float c = cosf(th), s = sinf(th);
  const float fx[4] = {-0.5f, 0.5f, 0.5f, -0.5f};
  const float fy[4] = {-0.5f, -0.5f, 0.5f, 0.5f};
#pragma unroll
  for (int k = 0; k < 4; ++k) {
    float dx = fx[k] * b[2], dy = fy[k] * b[3];
    cx[k] = b[0] + dx * c - dy * s;
    cy[k] = b[1] + dx * s + dy * c;
  }
}

__device__ float polyclip_area_dev(const float ax[4], const float ay[4],
                                   const float bx[4], const float by[4]) {
  float px[10], py[10];
  int n = 4;
  for (int k = 0; k < 4; ++k) { px[k] = ax[k]; py[k] = ay[k]; }
  for (int k = 0; k < 4; ++k) {                 // Sutherland-Hodgman, 4 edges
    int k1 = (k + 1) & 3;
    float ex = bx[k1] - bx[k], ey = by[k1] - by[k];
    float qx[10], qy[10];
    int m = 0;
    for (int v = 0; v < n; ++v) {
      int w = (v + 1 == n) ? 0 : (v + 1);
      float dc = ex * (py[v] - by[k]) - ey * (px[v] - bx[k]);
      float dn = ex * (py[w] - by[k]) - ey * (px[w] - bx[k]);
      bool ic = dc >= 0.0f, inx = dn >= 0.0f;
      if (ic) { qx[m] = px[v]; qy[m] = py[v]; ++m; }
      if (ic != inx) {
        float den = dc - dn;
        float safe = (fabsf(den) > 1e-9f) ? den : 1.0f;
        float t = dc / safe;
        qx[m] = px[v] + t * (px[w] - px[v]);
        qy[m] = py[v] + t * (py[w] - py[v]);
        ++m;
      }
    }
    n = m;
    for (int v = 0; v < n; ++v) { px[v] = qx[v]; py[v] = qy[v]; }
    if (n == 0) break;
  }
  if (n < 3) return 0.0f;
  float area = 0.0f;
  for (int v = 0; v < n; ++v) {
    int w = (v + 1 == n) ? 0 : (v + 1);
    area += px[v] * py[w] - px[w] * py[v];
  }
  return 0.5f * fabsf(area);
}

// ---------- stage 4b: exact rotated IoU -> suppression bitmask --------------
// One block per (row i, image). Box table staged into LDS via CDNA5 async
// global->LDS loads. Each wave ballots 32 columns into one mask word.
__global__ __launch_bounds__(256) void iou_kernel(
    const float* __restrict__ boxes, const u32* __restrict__ pmask,
    u32* __restrict__ smask) {
  __shared__ float sb[PRE * 5];                   // 20 KB of 320 KB LDS
  const int i   = blockIdx.x;
  const int img = blockIdx.y;
  const float* gb = boxes + (size_t)img * PRE * 5;

  // Async DMA the whole box table into LDS (ASYNCcnt path, gfx1250).
  for (int k = threadIdx.x; k < PRE * 5; k += 256) {
    unsigned lds_off = (unsigned)(k * 4u);        // sb allocated at LDS offset 0
    u64 gaddr = (u64)(const void*)(gb + k);
    asm volatile("global_load_async_to_lds_b32 %0, %1, off"
                 :: "v"(lds_off), "v"(gaddr) : "memory");
  }
  asm volatile("s_wait_asynccnt 0" ::: "memory");
  __syncthreads();

  float bi[5];
#pragma unroll
  for (int c = 0; c < 5; ++c) bi[c] = sb[i * 5 + c];
  float rix[4], riy[4];
  rbox_corners_dev(bi, rix, riy);
  float areaI = bi[2] * bi[3];

  const int lane = threadIdx.x & 31;
  for (int it = 0; it < 4; ++it) {
    int j = it * 256 + threadIdx.x;
    bool sup = false;
    u32 pw = pmask[((size_t)(img * PRE + i)) * 32 + (j >> 5)];
    if ((pw >> (j & 31)) & 1u) {
      float bj[5];
#pragma unroll
      for (int c = 0; c < 5; ++c) bj[c] = sb[j * 5 + c];
      float cjx[4], cjy[4];
      rbox_corners_dev(bj, cjx, cjy);
      float inter = polyclip_area_dev(rix, riy, cjx, cjy);
      float uni = areaI + bj[2] * bj[3] - inter;
      sup = (inter / fmaxf(uni, 1e-9f)) > NMS_T;
    }
    u32 bw = __builtin_amdgcn_ballot_w32(sup);
    if (lane == 0) smask[((size_t)(img * PRE + i)) * 32 + (j >> 5)] = bw;
  }
}

// ---------- stage 5: greedy NMS, single wave, state in registers ------------
__global__ __launch_bounds__(32) void nms_kernel(const u32* __restrict__ smask,
                                                 const int* __restrict__ valid,
                                                 u32* __restrict__ keepw) {
  const int img = blockIdx.x;
  const int lane = threadIdx.x;                   // owns bits [lane*32, +32)
  u32 vw = 0;
  for (int b = 0; b < 32; ++b)
    vw |= (valid[img * PRE + lane * 32 + b] ? 1u : 0u) << b;
  u32 sup = 0, keep = 0;
  for (int i = 0; i < PRE; ++i) {
    int wi = i >> 5, bi = i & 31;
    u32 supw = (u32)__builtin_amdgcn_readlane((int)sup, wi);  // uniform lane id
    u32 vww  = (u32)__builtin_amdgcn_readlane((int)vw, wi);
    bool is_keep = ((vww >> bi) & 1u) && !((supw >> bi) & 1u);
    if (is_keep) {                                // wave-uniform branch
      if (lane == wi) keep |= 1u << bi;
      u32 row = smask[((size_t)(img * PRE + i)) * 32 + lane];
      if (lane < wi) row = 0;
      else if (lane == wi) {
        u32 lowmask = (bi == 31) ? 0xFFFFFFFFu : ((1u << (bi + 1)) - 1u);
        row &= ~lowmask;                          // only j > i suppress
      }
      sup |= row;
    }
  }
  keepw[img * 32 + lane] = keep;
}

// ---------- stage 6: final top-256 (LDS bitonic) + output -------------------
__global__ __launch_bounds__(1024) void final_kernel(
    const float* __restrict__ boxes, const float* __restrict__ scores,
    const u32* __restrict__ keepw, float* __restrict__ out) {
  __shared__ u64 key[PRE];
  const int img = blockIdx.x;
  const int t = threadIdx.x;
  bool kp = (keepw[img * 32 + (t >> 5)] >> (t & 31)) & 1u;
  float sel = kp ? scores[img * PRE + t] : -1e30f;
  key[t] = ((u64)f2sort(sel) << 32) | (u64)(0xFFFFFFFFu - (u32)t);
  __syncthreads();
  for (int k = 2; k <= PRE; k <<= 1) {
    for (int j = k >> 1; j > 0; j >>= 1) {
      int ixj = t ^ j;
      if (ixj > t) {
        u64 a = key[t], b = key[ixj];
        bool up = (t & k) == 0;                   // descending overall
        if (up ? (a < b) : (a > b)) { key[t] = b; key[ixj] = a; }
      }
      __syncthreads();
    }
  }
  if (t < POST) {
    u64 kk = key[t];
    u32 orig = 0xFFFFFFFFu - (u32)(kk & 0xFFFFFFFFu);
    float sc = sort2f((u32)(kk >> 32));
    bool fv = (keepw[img * 32 + (orig >> 5)] >> (orig & 31)) & 1u;
    float* o = out + (size_t)(img * POST + t) * 6;
    if (fv) {
      const float* B = boxes + (size_t)(img * PRE + orig) * 5;
#pragma unroll
      for (int c = 0; c < 5; ++c) o[c] = B[c];
      o[5] = sc;
    } else {
#pragma unroll
      for (int c = 0; c < 6; ++c) o[c] = 0.0f;
    }
  }
}

// ---------- host orchestration ---------------------------------------------
extern "C" void kernel_launch(void* const* d_in, const int* in_sizes, int n_in,
                              void* d_out, int out_size, void* d_ws,
                              size_t ws_size, hipStream_t stream) {
  (void)in_sizes; (void)n_in; (void)out_size; (void)ws_size;
  const float* obj = (const float*)d_in[0];   // (2,1,256,256)
  const float* reg = (const float*)d_in[1];   // (2,5,256,256)
  float* out = (float*)d_out;                 // (2,256,6)

  char* ws = (char*)d_ws;
  u64* keys     = (u64*)ws;                               // 1 MB
  float* boxes  = (float*)(ws + (1u << 20));              // 40 KB
  float* scores = boxes + NIMG * PRE * 5;                 // 8 KB
  int* valid    = (int*)(scores + NIMG * PRE);            // 8 KB
  u32* pmask    = (u32*)(valid + NIMG * PRE);             // 256 KB
  u32* smask    = pmask + NIMG * PRE * 32;                // 256 KB
  u32* keepw    = smask + NIMG * PRE * 32;                // 256 B

  const int total = NIMG * HW;
  build_keys_kernel<<<total / 256, 256, 0, stream>>>(obj, keys);
  for (int k = 2; k <= HW; k <<= 1)
    for (int j = k >> 1; j > 0; j >>= 1)
      bitonic_step_kernel<<<total / 256, 256, 0, stream>>>(keys, k, j);

  decode_kernel<<<NIMG, PRE, 0, stream>>>(keys, reg, boxes, scores, valid);

  dim3 qg(32, 64, NIMG);
  quickreject_kernel<<<qg, 32, 0, stream>>>(boxes, pmask);

  dim3 ig(PRE, NIMG);
  iou_kernel<<<ig, 256, 0, stream>>>(boxes, pmask, smask);

  nms_kernel<<<NIMG, 32, 0, stream>>>(smask, valid, keepw);
  final_kernel<<<NIMG, PRE, 0, stream>>>(boxes, scores, keepw, out);
}